// GEPSConv2D_64098091925940
// MI455X (gfx1250) — compile-verified
//
#include <hip/hip_runtime.h>

typedef __attribute__((ext_vector_type(2))) float v2f;
typedef __attribute__((ext_vector_type(8))) float v8f;
typedef __attribute__((address_space(3))) float f32_lds;

#define B_SZ   16
#define C_IN   32
#define C_OUT  32
#define HDIM   128
#define WDIM   128
#define CODE   2
#define KTOT   (C_IN * 3 * 3)          // 288, unfold order kk = i*9 + k*3 + l

// LDS x-tile: [i][kr][136], image data at w+4 (w=0..127), halo at 3 and 132.
#define XSTR_K  136
#define XSTR_I  (4 * XSTR_K)           // 544
// float offset of unfold element kk at column shift (l-1), halo-padded (no wrap math)
#define OFF(kk) (((kk) / 9) * XSTR_I + ((((kk) % 9) / 3) * XSTR_K) + ((kk) % 3))

// ---- CDNA5 async memory->LDS copies (ASYNCcnt-tracked, bypass VGPRs) ----
__device__ __forceinline__ void async_lds_b128(f32_lds* lds, const float* g) {
    asm volatile("global_load_async_to_lds_b128 %0, %1, off"
                 :: "v"(lds), "v"(g) : "memory");
}
__device__ __forceinline__ void async_lds_b32(f32_lds* lds, const float* g) {
    asm volatile("global_load_async_to_lds_b32 %0, %1, off"
                 :: "v"(lds), "v"(g) : "memory");
}
__device__ __forceinline__ void wait_async0() {
    asm volatile("s_wait_asynccnt 0x0" ::: "memory");
}

// ---------------- kernel 1: per-sample weight synthesis ----------------
__global__ __launch_bounds__(256) void synth_weights(
    const float* __restrict__ weight, const float* __restrict__ codes,
    const float* __restrict__ A, const float* __restrict__ Bm,
    float* __restrict__ wcomb)
{
    int idx = blockIdx.x * 256 + threadIdx.x;   // exactly B_SZ*C_OUT*KTOT threads
    int col = idx % KTOT;
    int o   = (idx / KTOT) % C_OUT;
    int b   = idx / (KTOT * C_OUT);
    int i   = col / 9;
    int kl  = col % 9;

    float acc = weight[(o * C_IN + i) * 9 + kl];
#pragma unroll
    for (int r = 0; r < CODE; ++r) {
        float t = 0.f;
#pragma unroll
        for (int c = 0; c < CODE; ++c)
            t += A[(i * CODE + c) * 9 + kl] * codes[(b * CODE + c) * CODE + r];
        acc += t * Bm[(o * CODE + r) * 9 + kl];
    }
    wcomb[(b * C_OUT + o) * KTOT + col] = acc;
}

// ---------------- kernel 2: implicit-GEMM circular conv via WMMA f32 ----------------
// One block per (sample, 2 image rows). 8 waves; each wave computes a 32x32 tile
// (4 accumulators), reusing A fragments across two 16-column B tiles.
__global__ __launch_bounds__(256) void hconv_wmma(
    const float* __restrict__ x, const float* __restrict__ wcomb,
    const float* __restrict__ codes, const float* __restrict__ bias,
    const float* __restrict__ bias_ctx, float* __restrict__ out)
{
    __shared__ __align__(16) float wlds[C_OUT * KTOT];      // 36 KB: W[32][288]
    __shared__ __align__(16) float xlds[C_IN * XSTR_I];     // 68 KB: x[i][kr][136], kr=h0-1..h0+2

    const int tid = threadIdx.x;
    const int b   = blockIdx.x >> 6;            // / (HDIM/2)
    const int h0  = (blockIdx.x & 63) << 1;     // first of two output rows

    // ---- stage synthesized weights: async memory->LDS, 16B granules ----
    const float* wsrc = wcomb + (size_t)b * (C_OUT * KTOT);
    for (int t = tid; t < (C_OUT * KTOT) / 4; t += 256)     // 2304 = 9*256
        async_lds_b128((f32_lds*)(wlds + t * 4), wsrc + t * 4);

    // ---- stage 4 wrapped input rows for all channels (main body) ----
    for (int t = tid; t < C_IN * 4 * (WDIM / 4); t += 256) { // 4096 = 16*256
        int i   = t >> 7;           // /(4*32)
        int rem = t & 127;
        int kr  = rem >> 5;
        int w4  = rem & 31;
        int hr  = (h0 + kr + HDIM - 1) & (HDIM - 1);
        async_lds_b128((f32_lds*)(xlds + i * XSTR_I + kr * XSTR_K + 4 + w4 * 4),
                       x + ((size_t)(b * C_IN + i) * HDIM + hr) * WDIM + w4 * 4);
    }
    // ---- halo cells: exactly 256 single-float fills (32 ch x 4 rows x 2 sides) ----
    {
        int i    = tid >> 3;
        int rem  = tid & 7;
        int kr   = rem >> 1;
        int side = rem & 1;
        int hr   = (h0 + kr + HDIM - 1) & (HDIM - 1);
        const float* srow = x + ((size_t)(b * C_IN + i) * HDIM + hr) * WDIM;
        async_lds_b32((f32_lds*)(xlds + i * XSTR_I + kr * XSTR_K + (side ? 132 : 3)),
                      srow + (side ? 0 : 127));
    }
    wait_async0();          // this wave's async copies complete
    __syncthreads();        // all waves' copies visible

    const int lane    = tid & 31;
    const int wv      = tid >> 5;
    const int halfsel = lane >> 4;      // lane-half: K pair {0,1} vs {2,3}
    const int row     = lane & 15;
    const int hrow    = wv >> 2;        // which of the 2 image rows this wave owns
    const int col0    = ((wv & 3) << 5) + row;              // first N-tile column
    const int h       = h0 + hrow;

    // loop-invariant bases: A via DS offset immediates, B via one add per element
    const int abase = row * KTOT + (halfsel << 1);
    const int xbase = hrow * XSTR_K + col0 + 4 - 1;         // (l-1) shift folded: OFF has +l

    v8f acc00 = {}, acc01 = {}, acc10 = {}, acc11 = {};

#pragma unroll
    for (int kk0 = 0; kk0 < KTOT; kk0 += 4) {
        // A fragments (ds_load_b64 w/ literal offsets; no per-iter VALU)
        v2f a0, a1;
        a0.x = wlds[abase + kk0];
        a0.y = wlds[abase + kk0 + 1];
        a1.x = wlds[abase + 16 * KTOT + kk0];
        a1.y = wlds[abase + 16 * KTOT + kk0 + 1];

        // B fragments: one cndmask-of-literals + one add per K element;
        // second N-tile is a +64B DS offset immediate.
        int s0 = halfsel ? OFF(kk0 + 2) : OFF(kk0 + 0);
        int s1 = halfsel ? OFF(kk0 + 3) : OFF(kk0 + 1);
        v2f b0, b1;
        b0.x = xlds[xbase + s0];
        b1.x = xlds[xbase + s0 + 16];
        b0.y = xlds[xbase + s1];
        b1.y = xlds[xbase + s1 + 16];

        acc00 = __builtin_amdgcn_wmma_f32_16x16x4_f32(false, a0, false, b0,
                                                      (short)0, acc00, false, false);
        acc01 = __builtin_amdgcn_wmma_f32_16x16x4_f32(false, a0, false, b1,
                                                      (short)0, acc01, false, false);
        acc10 = __builtin_amdgcn_wmma_f32_16x16x4_f32(false, a1, false, b0,
                                                      (short)0, acc10, false, false);
        acc11 = __builtin_amdgcn_wmma_f32_16x16x4_f32(false, a1, false, b1,
                                                      (short)0, acc11, false, false);
    }

    // per-sample bias: bias[o] + codes[b,0,0]*bc[0][o] + codes[b,1,1]*bc[1][o]
    float d0 = codes[b * (CODE * CODE) + 0];
    float d1 = codes[b * (CODE * CODE) + 3];

#pragma unroll
    for (int r = 0; r < 8; ++r) {
        int o0 = r + (halfsel << 3);    // C/D layout: VGPR r -> M = r + 8*lanehalf
        int o1 = o0 + 16;
        float cb0 = bias[o0] + d0 * bias_ctx[o0] + d1 * bias_ctx[C_OUT + o0];
        float cb1 = bias[o1] + d0 * bias_ctx[o1] + d1 * bias_ctx[C_OUT + o1];
        size_t p0 = (((size_t)b * C_OUT + o0) * HDIM + h) * WDIM;
        size_t p1 = (((size_t)b * C_OUT + o1) * HDIM + h) * WDIM;
        out[p0 + col0]      = acc00[r] + cb0;
        out[p0 + col0 + 16] = acc01[r] + cb0;
        out[p1 + col0]      = acc10[r] + cb1;
        out[p1 + col0 + 16] = acc11[r] + cb1;
    }
}

extern "C" void kernel_launch(void* const* d_in, const int* in_sizes, int n_in,
                              void* d_out, int out_size, void* d_ws, size_t ws_size,
                              hipStream_t stream) {
    const float* x        = (const float*)d_in[0];
    const float* codes    = (const float*)d_in[1];
    const float* weight   = (const float*)d_in[2];
    const float* A        = (const float*)d_in[3];
    const float* Bm       = (const float*)d_in[4];
    const float* bias     = (const float*)d_in[5];
    const float* bias_ctx = (const float*)d_in[6];
    float* out   = (float*)d_out;
    float* wcomb = (float*)d_ws;   // 16*32*288 floats = 576 KB scratch

    synth_weights<<<(B_SZ * C_OUT * KTOT) / 256, 256, 0, stream>>>(
        weight, codes, A, Bm, wcomb);
    hconv_wmma<<<B_SZ * (HDIM / 2), 256, 0, stream>>>(
        x, wcomb, codes, bias, bias_ctx, out);
}